// DistributedPagedAttention_20727512171094
// MI455X (gfx1250) — compile-verified
//
#include <hip/hip_runtime.h>
#include <math.h>

typedef __attribute__((ext_vector_type(2))) float v2f;
typedef __attribute__((ext_vector_type(4))) float v4f;
typedef __attribute__((ext_vector_type(8))) float v8f;
typedef __attribute__((ext_vector_type(4))) int   v4i;

#define AS1 __attribute__((address_space(1)))
#define AS3 __attribute__((address_space(3)))

#define W_RANKS 4
#define NUM_HEADS 32
#define HEAD_SIZE 128
#define NUM_KV_HEADS 8
#define QPG 4                 // queries per kv head (GQA group)
#define BLOCK_SZ 16
#define NUM_CACHE_BLOCKS 1024
#define MAX_BLOCKS 32
#define NUM_SEQS 32
#define ATT_SCALE 0.08838834764831845f   // 1/sqrt(128)
#define NEG_INF_F (-1e30f)

#define KSTRIDE 132           // padded LDS row stride (floats): 132 % 64 == 4
                              // -> K-fragment reads land on distinct banks

#if __has_builtin(__builtin_amdgcn_global_load_async_to_lds_b128)
#define HAVE_ASYNC_LDS 1
#else
#define HAVE_ASYNC_LDS 0
#endif

#if __has_builtin(__builtin_amdgcn_s_wait_asynccnt)
#define WAIT_ASYNC(n) __builtin_amdgcn_s_wait_asynccnt(n)
#else
#define WAIT_ASYNC(n) asm volatile("s_wait_asynccnt %0" ::"i"(n) : "memory")
#endif

// Copy one 16x128 fp32 tile (contiguous 8KB) into padded LDS rows.
// 16 x (32 lanes x 16B) = 8KB; lane <-> consecutive 16B => fully coalesced.
__device__ __forceinline__ void tile_to_lds(const float* __restrict__ g,
                                            float* l, int lane)
{
#pragma unroll
    for (int j = 0; j < 16; ++j) {
        const float* gs = g + j * HEAD_SIZE + lane * 4;
        float*       ld = l + j * KSTRIDE   + lane * 4;
#if HAVE_ASYNC_LDS
        __builtin_amdgcn_global_load_async_to_lds_b128(
            (AS1 v4i*)gs, (AS3 v4i*)ld, 0, 0);
#else
        *(v4f*)ld = *(const v4f*)gs;
#endif
    }
}

// ---------------------------------------------------------------------------
// Kernel 1: per-(rank, seq, kv-head) paged attention with online softmax.
// One wave32 per (w,s,g); K/V tiles double-buffered in LDS via async copies.
// ---------------------------------------------------------------------------
__global__ __launch_bounds__(32)
void dpa_partial_kernel(const float* __restrict__ query,
                        const float* __restrict__ key_cache,
                        const float* __restrict__ value_cache,
                        const int*   __restrict__ block_tables,
                        const int*   __restrict__ context_lens,
                        float* __restrict__ part_out,   // [W,S,G,QPG,D]
                        float* __restrict__ part_m,     // [W,S,G,QPG]
                        float* __restrict__ part_l)     // [W,S,G,QPG]
{
    __shared__ float k_lds[2][16 * KSTRIDE];
    __shared__ float v_lds[2][16 * KSTRIDE];
    __shared__ float pt[16][16];   // staged P^T tile: [token][query]

    const int lane = threadIdx.x;
    const int sg = blockIdx.x % (NUM_SEQS * NUM_KV_HEADS);
    const int w  = blockIdx.x / (NUM_SEQS * NUM_KV_HEADS);
    const int s  = sg / NUM_KV_HEADS;
    const int g  = sg % NUM_KV_HEADS;

    const int half = lane >> 4;       // 0 => lanes 0-15, 1 => lanes 16-31
    const int l15  = lane & 15;
    const int koff = 2 * half;        // K-subindex offset inside a 4-chunk

    const int  ctx  = context_lens[w * NUM_SEQS + s];
    const int  nblk = (ctx + BLOCK_SZ - 1) / BLOCK_SZ;
    const int* bt   = block_tables + ((size_t)w * NUM_SEQS + s) * MAX_BLOCKS;

    const size_t head_base =
        (((size_t)w * NUM_CACHE_BLOCKS) * NUM_KV_HEADS + g) * BLOCK_SZ * HEAD_SIZE;
    const size_t blk_stride = (size_t)NUM_KV_HEADS * BLOCK_SZ * HEAD_SIZE;

    // ---- kick off async staging of block 0 while we load Q ----------------
    {
        const size_t off0 = head_base + (size_t)bt[0] * blk_stride;
        tile_to_lds(key_cache   + off0, &k_lds[0][0], lane);
        tile_to_lds(value_cache + off0, &v_lds[0][0], lane);
    }

    // ---- Q as WMMA A-fragments: 32 chunks of 16x4 covering D=128 ----------
    // A layout (32b, 16x4): lanes 0-15 row M=lane, K=0,1; lanes 16-31 K=2,3.
    v2f qa[32];
    {
        const float* qp = query + (size_t)s * (NUM_HEADS * HEAD_SIZE)
                                + (size_t)(g * QPG) * HEAD_SIZE;
        const int row = l15;          // query row; rows >= QPG are padding
#pragma unroll
        for (int c = 0; c < 32; ++c) {
            if (row < QPG) {
                qa[c].x = qp[row * HEAD_SIZE + 4 * c + koff + 0];
                qa[c].y = qp[row * HEAD_SIZE + 4 * c + koff + 1];
            } else {
                qa[c].x = 0.0f;
                qa[c].y = 0.0f;
            }
        }
    }

    // ---- online-softmax state & transposed output accumulator -------------
    float mq[QPG], lq[QPG];
#pragma unroll
    for (int q = 0; q < QPG; ++q) { mq[q] = NEG_INF_F; lq[q] = 0.0f; }

    v8f o_acc[8];                     // out^T: 8 d-chunks of 16 rows each
#pragma unroll
    for (int dc = 0; dc < 8; ++dc)
#pragma unroll
        for (int r = 0; r < 8; ++r)
            o_acc[dc][r] = 0.0f;

    for (int tb = 0; tb < nblk; ++tb) {
        const int buf = tb & 1;

        // ---- prefetch next block's tiles into the other buffer -----------
        const bool have_next = (tb + 1) < nblk;
        if (have_next) {
            const size_t offn = head_base + (size_t)bt[tb + 1] * blk_stride;
            tile_to_lds(key_cache   + offn, &k_lds[buf ^ 1][0], lane);
            tile_to_lds(value_cache + offn, &v_lds[buf ^ 1][0], lane);
        }

        // wait for current buffer (32 newer async loads may stay in flight)
#if HAVE_ASYNC_LDS
        if (have_next) { WAIT_ASYNC(32); } else { WAIT_ASYNC(0); }
#endif
        __syncthreads();

        // ---- S = Q * K^T  (16x16 tile: rows=queries, cols=tokens) ---------
        v8f sacc = {};
        const float* kl = &k_lds[buf][l15 * KSTRIDE + koff];  // token = l15
#pragma unroll
        for (int c = 0; c < 32; ++c) {
            v2f kb;                  // B fragment of K^T (dims x tokens)
            kb.x = kl[4 * c + 0];
            kb.y = kl[4 * c + 1];
            sacc = __builtin_amdgcn_wmma_f32_16x16x4_f32(
                       false, qa[c], false, kb, (short)0, sacc, false, false);
        }

        // ---- masked online softmax over this 16-token tile ----------------
        const int  tok   = tb * BLOCK_SZ + l15;
        const bool valid = (tok < ctx) && (half == 0);
        float pvals[QPG];
        float alpha[QPG];
#pragma unroll
        for (int q = 0; q < QPG; ++q) {
            float lg = sacc[q] * ATT_SCALE;
            lg = valid ? lg : NEG_INF_F;
            // tile max across the 16-lane token group
            float tm = lg;
            tm = fmaxf(tm, __shfl_xor(tm, 1));
            tm = fmaxf(tm, __shfl_xor(tm, 2));
            tm = fmaxf(tm, __shfl_xor(tm, 4));
            tm = fmaxf(tm, __shfl_xor(tm, 8));
            float mn = fmaxf(mq[q], tm);
            float a  = __expf(mq[q] - mn);
            float p  = valid ? __expf(lg - mn) : 0.0f;
            float ps = p;
            ps += __shfl_xor(ps, 1);
            ps += __shfl_xor(ps, 2);
            ps += __shfl_xor(ps, 4);
            ps += __shfl_xor(ps, 8);
            lq[q] = lq[q] * a + ps;
            mq[q] = mn;
            // broadcast lower-half alpha to upper half (it owns D rows 8..15)
            alpha[q] = __shfl(a, l15);
            pvals[q] = p;
        }

        // per-lane rescale factor: this lane holds output column (query) l15
        const float al = (l15 == 0) ? alpha[0]
                       : (l15 == 1) ? alpha[1]
                       : (l15 == 2) ? alpha[2]
                       :              alpha[3];
#pragma unroll
        for (int dc = 0; dc < 8; ++dc)
#pragma unroll
            for (int r = 0; r < 8; ++r)
                o_acc[dc][r] *= al;

        // ---- stage P^T through LDS (transpose C-layout -> B-layout) -------
        __syncthreads();
        if (half == 0) {
            pt[l15][0] = pvals[0];
            pt[l15][1] = pvals[1];
            pt[l15][2] = pvals[2];
            pt[l15][3] = pvals[3];
        }
        __syncthreads();

        v2f pbf[4];                  // B fragments of P^T (tokens x queries)
#pragma unroll
        for (int c = 0; c < 4; ++c) {
            pbf[c].x = pt[4 * c + koff + 0][l15];
            pbf[c].y = pt[4 * c + koff + 1][l15];
        }

        // ---- out^T += V^T * P^T  (8 d-chunks x 4 token-chunks) ------------
        const float* vl = &v_lds[buf][l15];   // lane <-> head-dim (no conflicts)
#pragma unroll
        for (int dc = 0; dc < 8; ++dc) {
#pragma unroll
            for (int c = 0; c < 4; ++c) {
                v2f va;              // A fragment of V^T (dims x tokens)
                const int t0 = 4 * c + koff;
                va.x = vl[(t0 + 0) * KSTRIDE + dc * 16];
                va.y = vl[(t0 + 1) * KSTRIDE + dc * 16];
                o_acc[dc] = __builtin_amdgcn_wmma_f32_16x16x4_f32(
                                false, va, false, pbf[c], (short)0,
                                o_acc[dc], false, false);
            }
        }
        __syncthreads();   // protect LDS buffers before next iteration reuse
    }

    // ---- write per-rank partials ------------------------------------------
    // out^T D layout: VGPR r, lanes 0-15 -> d = dc*16 + r, lanes 16-31 -> +8,
    // column (query) = lane & 15.
    const int qcol = l15;
    if (qcol < QPG) {
        float* po = part_out + (size_t)blockIdx.x * (QPG * HEAD_SIZE)
                             + (size_t)qcol * HEAD_SIZE;
#pragma unroll
        for (int dc = 0; dc < 8; ++dc)
#pragma unroll
            for (int r = 0; r < 8; ++r)
                po[dc * 16 + half * 8 + r] = o_acc[dc][r];
    }
    if (lane == 0) {
#pragma unroll
        for (int q = 0; q < QPG; ++q) {
            part_m[(size_t)blockIdx.x * QPG + q] = mq[q];
            part_l[(size_t)blockIdx.x * QPG + q] = lq[q];
        }
    }
}

// ---------------------------------------------------------------------------
// Kernel 2: cross-rank log-sum-exp reduction.
// ---------------------------------------------------------------------------
__global__ __launch_bounds__(256)
void dpa_reduce_kernel(const float* __restrict__ part_out,
                       const float* __restrict__ part_m,
                       const float* __restrict__ part_l,
                       float* __restrict__ out)
{
    const int total = NUM_SEQS * NUM_KV_HEADS * QPG * HEAD_SIZE;
    const int idx = blockIdx.x * blockDim.x + threadIdx.x;
    if (idx >= total) return;

    const int d = idx & (HEAD_SIZE - 1);
    const int q = (idx / HEAD_SIZE) & (QPG - 1);
    const int g = (idx / (HEAD_SIZE * QPG)) % NUM_KV_HEADS;
    const int s = idx / (HEAD_SIZE * QPG * NUM_KV_HEADS);

    const int sgq      = (s * NUM_KV_HEADS + g) * QPG + q;
    const int stride_m = NUM_SEQS * NUM_KV_HEADS * QPG;  // per-rank stride

    float gmax = NEG_INF_F;
#pragma unroll
    for (int w = 0; w < W_RANKS; ++w)
        gmax = fmaxf(gmax, part_m[w * stride_m + sgq]);

    float num = 0.0f, den = 0.0f;
#pragma unroll
    for (int w = 0; w < W_RANKS; ++w) {
        const float e = __expf(part_m[w * stride_m + sgq] - gmax);
        den += part_l[w * stride_m + sgq] * e;
        num += part_out[((size_t)w * stride_m + sgq) * HEAD_SIZE + d] * e;
    }

    out[((size_t)s * NUM_HEADS + g * QPG + q) * HEAD_SIZE + d] = num / den;
}

// ---------------------------------------------------------------------------
extern "C" void kernel_launch(void* const* d_in, const int* in_sizes, int n_in,
                              void* d_out, int out_size, void* d_ws, size_t ws_size,
                              hipStream_t stream)
{
    const float* query       = (const float*)d_in[0];
    const float* key_cache   = (const float*)d_in[1];
    const float* value_cache = (const float*)d_in[2];
    const int*   block_tbl   = (const int*)d_in[3];
    const int*   ctx_lens    = (const int*)d_in[4];
    float*       out         = (float*)d_out;

    // workspace layout (floats)
    const size_t n_wsg   = (size_t)W_RANKS * NUM_SEQS * NUM_KV_HEADS; // 1024
    float* part_out = (float*)d_ws;                                   // n_wsg*512
    float* part_m   = part_out + n_wsg * QPG * HEAD_SIZE;             // n_wsg*4
    float* part_l   = part_m   + n_wsg * QPG;                         // n_wsg*4

    dpa_partial_kernel<<<dim3((unsigned)n_wsg), dim3(32), 0, stream>>>(
        query, key_cache, value_cache, block_tbl, ctx_lens,
        part_out, part_m, part_l);

    const int total   = NUM_SEQS * NUM_KV_HEADS * QPG * HEAD_SIZE;    // 131072
    const int threads = 256;
    dpa_reduce_kernel<<<dim3((total + threads - 1) / threads), dim3(threads),
                        0, stream>>>(part_out, part_m, part_l, out);
}